// ProgRnCLoss_53386443489453
// MI455X (gfx1250) — compile-verified
//
#include <hip/hip_runtime.h>

#define N 320
#define D 256

typedef __attribute__((ext_vector_type(2))) float v2f;
typedef __attribute__((ext_vector_type(8))) float v8f;

// ---------------------------------------------------------------------------
// Kernel 1: G = F * F^T via V_WMMA_F32_16X16X4_F32.
// One wave (32 lanes) computes one 16x16 tile of G, looping K in steps of 4.
// A-matrix 16x4 f32 layout: lanes 0-15 = rows M=0..15 holding K={k0,k0+1},
// lanes 16-31 = same rows holding K={k0+2,k0+3}.  B (4x16) mirrors with lanes
// indexing N.  Both tiles read rows of F, so the per-lane load is identical.
// ---------------------------------------------------------------------------
__global__ __launch_bounds__(32) void gram_wmma(const float* __restrict__ F,
                                                float* __restrict__ G) {
  const int tm   = blockIdx.x * 16;
  const int tn   = blockIdx.y * 16;
  const int lane = threadIdx.x;          // 0..31, wave32
  const int half = lane >> 4;            // 0 or 1
  const int l16  = lane & 15;
  const float* Arow = F + (tm + l16) * D + 2 * half;
  const float* Brow = F + (tn + l16) * D + 2 * half;
  v8f acc = {};
  for (int k0 = 0; k0 < D; k0 += 4) {
    v2f a = *(const v2f*)(Arow + k0);
    v2f b = *(const v2f*)(Brow + k0);
    // 8 args: (neg_a, A, neg_b, B, c_mod, C, reuse_a, reuse_b)
    acc = __builtin_amdgcn_wmma_f32_16x16x4_f32(
        false, a, false, b, (short)0, acc, false, false);
  }
  // C/D layout: VGPR r -> M = r + 8*half, N = l16
  const int n = tn + l16;
#pragma unroll
  for (int r = 0; r < 8; ++r) {
    G[(tm + r + 8 * half) * N + n] = acc[r];
  }
}

// ---------------------------------------------------------------------------
// Kernel 2: E[i][j] = exp(-sqrt(d2)/TEMP), d2 = G_ii + G_jj - 2 G_ij.
// Diagonal d2 is exactly 0 -> logit 0, so row-max (always 0) is skipped.
// ---------------------------------------------------------------------------
__global__ __launch_bounds__(256) void exp_logits_k(const float* __restrict__ G,
                                                    float* __restrict__ E) {
  int idx = blockIdx.x * 256 + threadIdx.x;
  if (idx >= N * N) return;
  int i = idx / N, j = idx % N;
  float d2 = G[i * N + i] + G[j * N + j] - 2.0f * G[idx];
  float lg = (d2 > 0.0f) ? (-0.5f * sqrtf(d2)) : 0.0f;  // TEMP = 2
  E[idx] = expf(lg);
}

// ---------------------------------------------------------------------------
// Kernel 3: per-anchor masked reduction.  One wave per (k,i); lanes stride j.
// The 7x7 {0,1,2} table row for (k,i) is wave-uniform -> packed into 2-bit
// fields of a scalar.  Deterministic xor-tree reduction (no atomics).
// ---------------------------------------------------------------------------
__global__ __launch_bounds__(256) void anchor_k(const float* __restrict__ G,
                                                const float* __restrict__ E,
                                                const float* __restrict__ t,
                                                const int* __restrict__ ev,
                                                float* __restrict__ P,
                                                float* __restrict__ V) {
  __shared__ float st[N];
  __shared__ int   se[N];
  for (int x = threadIdx.x; x < N; x += 256) { st[x] = t[x]; se[x] = ev[x]; }
  __syncthreads();

  const int k    = blockIdx.y;
  const int i    = blockIdx.x * 8 + (threadIdx.x >> 5);
  const int lane = threadIdx.x & 31;

  const float tk   = st[k];
  const float ti   = st[i];
  const float pldi = ti - tk;
  const float absi = fabsf(pldi);
  const int   sgn  = (pldi > 0.0f) ? 1 : ((pldi < 0.0f) ? -1 : 0);
  const int   cid  = (se[i] + 10 * se[k]) * sgn;

  int cidx;
  switch (cid) {
    case -11: cidx = 0; break; case -10: cidx = 1; break;
    case  -1: cidx = 2; break; case   0: cidx = 3; break;
    case   1: cidx = 4; break; case  10: cidx = 5; break;
    default:  cidx = 6; break;  // cid == 11
  }
#define PK(a,b,c,d,e,f,g) \
  ((unsigned)((a)|((b)<<2)|((c)<<4)|((d)<<6)|((e)<<8)|((f)<<10)|((g)<<12)))
  unsigned rowbits;
  switch (cidx) {
    case 0:  rowbits = PK(0,2,2,1,1,0,1); break;
    case 1:  rowbits = PK(0,2,2,2,2,2,2); break;
    case 2:  rowbits = PK(0,2,2,2,1,0,2); break;
    case 3:  rowbits = PK(0,2,2,2,2,2,2); break;
    case 4:  rowbits = PK(0,2,2,2,1,2,2); break;
    case 5:  rowbits = PK(0,2,2,2,2,0,2); break;
    default: rowbits = PK(0,2,2,1,1,0,1); break;
  }
#undef PK

  const bool izero = (i == k);
  float wsum = 0.0f, denom = 0.0f;
  const float* Ei = E + i * N;
  for (int j = lane; j < N; j += 32) {
    float pj = st[j] - tk;
    int td;
    if (pj > absi)        td = 3;
    else if (-pj > absi)  td = 1;
    else if (pj != 0.0f)  td = 2;
    else                  td = 0;
    if (j == i) td = 0;
    int neg    = td ? (se[j] * 10 + td) : 0;
    int vidx   = (neg > 3) ? (neg - 7) : neg;
    int mapped = (int)((rowbits >> (2 * vidx)) & 3u);
    float w = (mapped == 1) ? 1.0f : ((mapped == 2) ? 0.5f : 0.0f);
    if (izero || j == k) w = 0.0f;
    wsum  += w;
    denom += w * Ei[j];
  }
  for (int off = 16; off >= 1; off >>= 1) {
    wsum  += __shfl_xor(wsum, off, 32);
    denom += __shfl_xor(denom, off, 32);
  }
  if (lane == 0) {
    bool  valid = wsum > 0.0f;
    float d2 = G[i * N + i] + G[k * N + k] - 2.0f * G[i * N + k];
    float lg = (d2 > 0.0f) ? (-0.5f * sqrtf(d2)) : 0.0f;  // logits[i,k]
    float logp = lg - logf(valid ? denom : 1.0f);
    P[k * N + i] = valid ? logp : 0.0f;
    V[k * N + i] = valid ? 1.0f : 0.0f;
  }
}

// ---------------------------------------------------------------------------
// Kernel 4: loss_k = Np>0 ? -sum_i P / Np : 0 ; out = sum_k loss_k / #(Np>0)
// ---------------------------------------------------------------------------
__global__ __launch_bounds__(320) void finalize_k(const float* __restrict__ P,
                                                  const float* __restrict__ V,
                                                  float* __restrict__ out) {
  __shared__ float sl[N];
  __shared__ int   shp[N];
  const int k = threadIdx.x;  // 0..319
  float sp = 0.0f, nv = 0.0f;
  for (int i = 0; i < N; ++i) { sp += P[k * N + i]; nv += V[k * N + i]; }
  int np = (int)(nv + 0.5f);
  sl[k]  = (np > 0) ? (-sp / (float)np) : 0.0f;
  shp[k] = (np > 0) ? 1 : 0;
  __syncthreads();
  if (k == 0) {
    float s = 0.0f; int c = 0;
    for (int x = 0; x < N; ++x) { s += sl[x]; c += shp[x]; }
    out[0] = s / (float)c;
  }
}

extern "C" void kernel_launch(void* const* d_in, const int* in_sizes, int n_in,
                              void* d_out, int out_size, void* d_ws, size_t ws_size,
                              hipStream_t stream) {
  (void)in_sizes; (void)n_in; (void)out_size; (void)ws_size;
  const float* F  = (const float*)d_in[0];   // features (320,256) f32
  const float* t  = (const float*)d_in[1];   // y_times  (320,)    f32
  const int*   ev = (const int*)d_in[2];     // y_events (320,)    i32
  float* out = (float*)d_out;

  float* G = (float*)d_ws;        // 320*320 f32
  float* E = G + N * N;           // 320*320 f32
  float* P = E + N * N;           // 320*320 f32
  float* V = P + N * N;           // 320*320 f32  (total ~1.6 MB)

  gram_wmma  <<<dim3(N / 16, N / 16), 32,  0, stream>>>(F, G);
  exp_logits_k<<<(N * N + 255) / 256, 256, 0, stream>>>(G, E);
  anchor_k   <<<dim3(N / 8, N),       256, 0, stream>>>(G, E, t, ev, P, V);
  finalize_k <<<1,                    320, 0, stream>>>(P, V, out);
}